// GCN_48412871360961
// MI455X (gfx1250) — compile-verified
//
#include <hip/hip_runtime.h>

#define N_NODES 20000
#define N_EDGES 320000
#define D_IN    256
#define D_OUT   256

typedef __attribute__((ext_vector_type(2))) float v2f;
typedef __attribute__((ext_vector_type(8))) float v8f;

// ---------------------------------------------------------------------------
// 1) Init: zero the output accumulator, deg=1.0 (self-loop contribution)
// ---------------------------------------------------------------------------
__global__ __launch_bounds__(256) void gcn_init(float* __restrict__ out,
                                                float* __restrict__ deg) {
    int idx = blockIdx.x * blockDim.x + threadIdx.x;
    if (idx < N_NODES * D_OUT) out[idx] = 0.0f;
    if (idx < N_NODES)         deg[idx] = 1.0f;
}

// ---------------------------------------------------------------------------
// 2) Degree accumulation over edge destinations (A is int64 from JAX)
// ---------------------------------------------------------------------------
__global__ __launch_bounds__(256) void gcn_deg(const long long* __restrict__ A,
                                               float* __restrict__ deg) {
    int e = blockIdx.x * blockDim.x + threadIdx.x;
    if (e < N_EDGES) {
        int d = (int)A[N_EDGES + e];
        unsafeAtomicAdd(&deg[d], 1.0f);   // hardware global_atomic_add_f32
    }
}

// ---------------------------------------------------------------------------
// 3) dinv = rsqrt(deg)  (deg >= 1 always, guard matches reference anyway)
// ---------------------------------------------------------------------------
__global__ __launch_bounds__(256) void gcn_dinv(const float* __restrict__ deg,
                                                float* __restrict__ dinv) {
    int i = blockIdx.x * blockDim.x + threadIdx.x;
    if (i < N_NODES) {
        float d = deg[i];
        dinv[i] = (d > 0.0f) ? rsqrtf(d) : 0.0f;
    }
}

// ---------------------------------------------------------------------------
// 4) H = X @ W via V_WMMA_F32_16X16X4_F32. One wave per 16x16 tile.
//    grid: 2500 blocks * 8 waves = 20000 waves = 1250 (M tiles) * 16 (N tiles)
// ---------------------------------------------------------------------------
__global__ __launch_bounds__(256) void gcn_gemm(const float* __restrict__ X,
                                                const float* __restrict__ W,
                                                float* __restrict__ H) {
    const int lane = threadIdx.x & 31;
    const int wid  = blockIdx.x * (blockDim.x >> 5) + (threadIdx.x >> 5);
    const int tm   = wid >> 4;        // 0..1249
    const int tn   = wid & 15;        // 0..15
    const int m0   = tm << 4;
    const int n0   = tn << 4;
    const int half = lane >> 4;       // 0: K={0,1}, 1: K={2,3}
    const int l    = lane & 15;

    const float* __restrict__ Xrow = X + (size_t)(m0 + l) * D_IN;  // A row m0+l
    const float* __restrict__ Wcol = W + n0 + l;                   // B col n0+l

    v8f acc = {};
#pragma unroll 8
    for (int k0 = 0; k0 < D_IN; k0 += 4) {
        const int kk = k0 + (half << 1);
        v2f a, b;
        a.x = Xrow[kk + 0];
        a.y = Xrow[kk + 1];
        b.x = Wcol[(size_t)(kk + 0) * D_OUT];
        b.y = Wcol[(size_t)(kk + 1) * D_OUT];
        // 8 args: (neg_a, A, neg_b, B, c_mod, C, reuse_a, reuse_b)
        acc = __builtin_amdgcn_wmma_f32_16x16x4_f32(
                false, a, false, b, (short)0, acc, false, false);
    }

    // C/D layout: VGPR r -> row m0 + r + 8*half, col n0 + l
    float* __restrict__ Hcol = H + n0 + l;
#pragma unroll
    for (int r = 0; r < 8; ++r) {
        Hcol[(size_t)(m0 + r + (half << 3)) * D_OUT] = acc[r];
    }
}

// ---------------------------------------------------------------------------
// 5) Edge scatter: one wave per edge. Coalesced float4 gather of h[src],
//    scale by dinv[src]*dinv[dst], 256 f32 fadd atomics into out[dst].
// ---------------------------------------------------------------------------
__global__ __launch_bounds__(256) void gcn_scatter(const long long* __restrict__ A,
                                                   const float* __restrict__ H,
                                                   const float* __restrict__ dinv,
                                                   float* __restrict__ out) {
    const int lane = threadIdx.x & 31;
    const int e    = blockIdx.x * (blockDim.x >> 5) + (threadIdx.x >> 5);
    if (e >= N_EDGES) return;                       // wave-uniform guard
    const int   s   = (int)A[e];
    const int   d   = (int)A[N_EDGES + e];
    const float nrm = dinv[s] * dinv[d];

    const float4* __restrict__ hs = (const float4*)(H + (size_t)s * D_OUT);
    float4 h0 = hs[lane];        // channels [4*lane .. 4*lane+3]
    float4 h1 = hs[lane + 32];   // channels [128+4*lane ..]

    float* __restrict__ od = out + (size_t)d * D_OUT;
    const int c = lane << 2;
    unsafeAtomicAdd(od + c + 0,       h0.x * nrm);
    unsafeAtomicAdd(od + c + 1,       h0.y * nrm);
    unsafeAtomicAdd(od + c + 2,       h0.z * nrm);
    unsafeAtomicAdd(od + c + 3,       h0.w * nrm);
    unsafeAtomicAdd(od + c + 128,     h1.x * nrm);
    unsafeAtomicAdd(od + c + 129,     h1.y * nrm);
    unsafeAtomicAdd(od + c + 130,     h1.z * nrm);
    unsafeAtomicAdd(od + c + 131,     h1.w * nrm);
}

// ---------------------------------------------------------------------------
// 6) Finalize: out = relu(out + h[i]*dinv[i]^2 (self-loop) + b)
// ---------------------------------------------------------------------------
__global__ __launch_bounds__(256) void gcn_final(const float* __restrict__ H,
                                                 const float* __restrict__ dinv,
                                                 const float* __restrict__ b,
                                                 float* __restrict__ out) {
    int idx = blockIdx.x * blockDim.x + threadIdx.x;
    if (idx >= N_NODES * D_OUT) return;
    int i = idx >> 8;
    int c = idx & 255;
    float di = dinv[i];
    float v  = out[idx] + H[idx] * di * di + b[c];
    out[idx] = fmaxf(v, 0.0f);
}

// ---------------------------------------------------------------------------
extern "C" void kernel_launch(void* const* d_in, const int* in_sizes, int n_in,
                              void* d_out, int out_size, void* d_ws, size_t ws_size,
                              hipStream_t stream) {
    const long long* A = (const long long*)d_in[0];  // [2, N_EDGES] int64
    const float*     X = (const float*)d_in[1];      // [N_NODES, D_IN]
    const float*     W = (const float*)d_in[2];      // [D_IN, D_OUT]
    const float*     b = (const float*)d_in[3];      // [D_OUT]
    float*         out = (float*)d_out;              // [N_NODES, D_OUT]

    // Workspace layout: H [N*D_OUT] | deg [N] | dinv [N]   (~20.6 MB)
    float* H    = (float*)d_ws;
    float* deg  = H + (size_t)N_NODES * D_OUT;
    float* dinv = deg + N_NODES;

    const int ND = N_NODES * D_OUT;

    gcn_init<<<(ND + 255) / 256, 256, 0, stream>>>(out, deg);
    gcn_deg<<<(N_EDGES + 255) / 256, 256, 0, stream>>>(A, deg);
    gcn_dinv<<<(N_NODES + 255) / 256, 256, 0, stream>>>(deg, dinv);

    // 20000 tiles, 8 waves (256 threads) per block -> 2500 blocks exactly
    gcn_gemm<<<2500, 256, 0, stream>>>(X, W, H);

    // one wave per edge, 8 waves per block -> 40000 blocks exactly
    gcn_scatter<<<N_EDGES / 8, 256, 0, stream>>>(A, H, dinv, out);

    gcn_final<<<(ND + 255) / 256, 256, 0, stream>>>(H, dinv, b, out);
}